// TGCNCell_4363686772767
// MI455X (gfx1250) — compile-verified
//
#include <hip/hip_runtime.h>
#include <math.h>

// Problem constants (from reference)
#define NN 50000
#define NE 800000
// IN=128, OUT=64, CAT=192

typedef float v2f __attribute__((ext_vector_type(2)));
typedef float v8f __attribute__((ext_vector_type(8)));

// Native f32 global atomic add (avoids CAS-loop fallback of atomicAdd(float)).
__device__ __forceinline__ void atomic_add_f32(float* p, float v) {
    asm volatile("global_atomic_add_f32 %0, %1, off" :: "v"(p), "v"(v) : "memory");
}

// ---------------- degree / normalization ----------------
__global__ void k_deg_init(float* deg) {
    int i = blockIdx.x * blockDim.x + threadIdx.x;
    if (i < NN) deg[i] = 1.0f;   // self-loop weight
}

__global__ void k_deg_edges(const long long* __restrict__ col,
                            const float* __restrict__ w, float* deg) {
    int e = blockIdx.x * blockDim.x + threadIdx.x;
    if (e < NE) atomic_add_f32(&deg[col[e]], w[e]);
}

__global__ void k_dinv(float* deg) {
    int i = blockIdx.x * blockDim.x + threadIdx.x;
    if (i < NN) {
        float d = deg[i];
        deg[i] = (d > 0.0f) ? rsqrtf(d) : 0.0f;
    }
}

// ---------------- dense GEMM: [X | A2] (N x 192) @ W (192 x ncols) ----------------
// ncols = 128 -> B0=W_z, B1=W_r fused side by side (8 waves); ncols = 64 -> B0=W_h (4 waves).
// One wave computes one 16x16 output tile via 48x V_WMMA_F32_16X16X4_F32.
__global__ void k_gemm_cat(const float* __restrict__ X, const float* __restrict__ A2,
                           const float* __restrict__ B0, const float* __restrict__ B1,
                           float* __restrict__ out, int ncols) {
    __shared__ float tileA[16 * 196];          // 16 rows x 192 cols, pitch 196 (bank-clean)
    const int lane = threadIdx.x & 31;
    const int wave = threadIdx.x >> 5;
    const int n0   = wave << 4;                // this wave's output column tile
    const int ct   = lane & 15;                // column within tile
    const int koff = (lane >> 4) << 1;         // K sub-offset per ISA A/B f32 layout

    // Pre-load all 48 B fragments (weights are tiny; amortized over row tiles).
    const float* B  = (n0 >= 64) ? B1 : B0;
    const int bcol  = ((n0 >= 64) ? (n0 - 64) : n0) + ct;
    v2f bfrag[48];
#pragma unroll
    for (int kk = 0; kk < 48; ++kk) {
        const int kb = kk * 4 + koff;
        v2f b;
        b.x = B[kb * 64 + bcol];
        b.y = B[(kb + 1) * 64 + bcol];
        bfrag[kk] = b;
    }

    const int nRowTiles = NN / 16;             // 3125, exact
    for (int rt = blockIdx.x; rt < nRowTiles; rt += gridDim.x) {
        const int m0 = rt << 4;
        // Stage A tile (concat of X cols 0..127 and A2 cols 0..63) into LDS.
        for (int idx = threadIdx.x; idx < 16 * 192; idx += blockDim.x) {
            const int r  = idx / 192;
            const int c  = idx - r * 192;
            const int gr = m0 + r;
            tileA[r * 196 + c] = (c < 128) ? X[gr * 128 + c] : A2[gr * 64 + (c - 128)];
        }
        __syncthreads();

        v8f acc = {};
        const float* arow = &tileA[ct * 196 + koff];
#pragma unroll
        for (int kk = 0; kk < 48; ++kk) {
            v2f a = *(const v2f*)(arow + kk * 4);
            acc = __builtin_amdgcn_wmma_f32_16x16x4_f32(
                false, a, false, bfrag[kk], (short)0, acc, false, false);
        }

        // Store C per ISA layout: VGPR r -> row m0+r (lanes 0-15) / m0+8+r (lanes 16-31).
        const int nc    = n0 + ct;
        const int rbase = m0 + ((lane >> 4) << 3);
#pragma unroll
        for (int r = 0; r < 8; ++r)
            out[(rbase + r) * ncols + nc] = acc[r];
        __syncthreads();
    }
}

// ---------------- aggregate init with self-loop term: agg = H * dinv^2 ----------------
__global__ void k_init_agg(const float* __restrict__ H, const float* __restrict__ dinv,
                           float* __restrict__ agg, int shift, int total) {
    for (int idx = blockIdx.x * blockDim.x + threadIdx.x; idx < total;
         idx += gridDim.x * blockDim.x) {
        const float di = dinv[idx >> shift];
        agg[idx] = H[idx] * di * di;
    }
}

// ---------------- edge scatter: agg[col] += H[row] * norm ----------------
// One wave per edge; 128 channels -> float4 per lane.
__global__ void k_edge128(const long long* __restrict__ row, const long long* __restrict__ col,
                          const float* __restrict__ w, const float* __restrict__ dinv,
                          const float* __restrict__ H, float* __restrict__ agg) {
    const int gw = (blockIdx.x * blockDim.x + threadIdx.x) >> 5;
    if (gw >= NE) return;
    const int lane = threadIdx.x & 31;
    const long long r0 = row[gw];
    const long long c0 = col[gw];
    const float nrm = dinv[r0] * w[gw] * dinv[c0];
    const float4 v = *(const float4*)(H + r0 * 128 + lane * 4);
    float* dst = agg + c0 * 128 + lane * 4;
    atomic_add_f32(dst + 0, v.x * nrm);
    atomic_add_f32(dst + 1, v.y * nrm);
    atomic_add_f32(dst + 2, v.z * nrm);
    atomic_add_f32(dst + 3, v.w * nrm);
}

// 64 channels -> float2 per lane.
__global__ void k_edge64(const long long* __restrict__ row, const long long* __restrict__ col,
                         const float* __restrict__ w, const float* __restrict__ dinv,
                         const float* __restrict__ H, float* __restrict__ agg) {
    const int gw = (blockIdx.x * blockDim.x + threadIdx.x) >> 5;
    if (gw >= NE) return;
    const int lane = threadIdx.x & 31;
    const long long r0 = row[gw];
    const long long c0 = col[gw];
    const float nrm = dinv[r0] * w[gw] * dinv[c0];
    const float2 v = *(const float2*)(H + r0 * 64 + lane * 2);
    float* dst = agg + c0 * 64 + lane * 2;
    atomic_add_f32(dst + 0, v.x * nrm);
    atomic_add_f32(dst + 1, v.y * nrm);
}

// ---------------- z/r gates: z = sig(aggZ+bz); rh = sig(aggR+br) * hidden ----------------
__global__ void k_rgate(const float* __restrict__ agg, const float* __restrict__ bz,
                        const float* __restrict__ br, const float* __restrict__ hid,
                        float* __restrict__ z, float* __restrict__ rh) {
    for (int idx = blockIdx.x * blockDim.x + threadIdx.x; idx < NN * 64;
         idx += gridDim.x * blockDim.x) {
        const int n = idx >> 6;
        const int c = idx & 63;
        const float zv = 1.0f / (1.0f + expf(-(agg[n * 128 + c] + bz[c])));
        const float rv = 1.0f / (1.0f + expf(-(agg[n * 128 + 64 + c] + br[c])));
        z[idx]  = zv;
        rh[idx] = rv * hid[idx];
    }
}

// ---------------- final GRU blend: out = z*h + (1-z)*tanh(aggH + bh) ----------------
__global__ void k_final(const float* __restrict__ agg, const float* __restrict__ bh,
                        const float* __restrict__ z, const float* __restrict__ hid,
                        float* __restrict__ out) {
    for (int idx = blockIdx.x * blockDim.x + threadIdx.x; idx < NN * 64;
         idx += gridDim.x * blockDim.x) {
        const int c = idx & 63;
        const float hc = tanhf(agg[idx] + bh[c]);
        const float zv = z[idx];
        out[idx] = zv * hid[idx] + (1.0f - zv) * hc;
    }
}

extern "C" void kernel_launch(void* const* d_in, const int* in_sizes, int n_in,
                              void* d_out, int out_size, void* d_ws, size_t ws_size,
                              hipStream_t stream) {
    (void)in_sizes; (void)n_in; (void)out_size; (void)ws_size;
    const float*      x   = (const float*)d_in[0];
    const long long*  ei  = (const long long*)d_in[1];  // int64 (2, E)
    const float*      w   = (const float*)d_in[2];
    const float*      hid = (const float*)d_in[3];
    const float*      Wz  = (const float*)d_in[4];
    const float*      bz  = (const float*)d_in[5];
    const float*      Wr  = (const float*)d_in[6];
    const float*      br  = (const float*)d_in[7];
    const float*      Wh  = (const float*)d_in[8];
    const float*      bh  = (const float*)d_in[9];
    float*            out = (float*)d_out;

    // Workspace layout (floats): dinv | H (N*128, reused N*64) | agg (N*128, reused N*64) | z | rh
    float* ws   = (float*)d_ws;
    float* dinv = ws;
    float* H    = ws + 50048;                  // 16B-aligned offset
    float* agg  = H + NN * 128;
    float* z    = agg + NN * 128;
    float* rh   = z + NN * 64;

    const long long* row = ei;
    const long long* col = ei + NE;

    // 1) GCN normalization: deg -> dinv
    k_deg_init<<<(NN + 255) / 256, 256, 0, stream>>>(dinv);
    k_deg_edges<<<(NE + 255) / 256, 256, 0, stream>>>(col, w, dinv);
    k_dinv<<<(NN + 255) / 256, 256, 0, stream>>>(dinv);

    // 2) Fused z|r GEMM: H(N,128) = [x|hid] @ [Wz|Wr]
    k_gemm_cat<<<480, 256, 0, stream>>>(x, hid, Wz, Wr, H, 128);

    // 3) Aggregate (self-loops folded into init) for z and r together
    k_init_agg<<<2048, 256, 0, stream>>>(H, dinv, agg, 7, NN * 128);
    k_edge128<<<NE / 8, 256, 0, stream>>>(row, col, w, dinv, H, agg);

    // 4) Gates: z and r*h
    k_rgate<<<2048, 256, 0, stream>>>(agg, bz, br, hid, z, rh);

    // 5) Candidate GEMM: H(N,64) = [x | r*h] @ Wh   (reuses H buffer)
    k_gemm_cat<<<480, 128, 0, stream>>>(x, rh, Wh, nullptr, H, 64);

    // 6) Aggregate candidate (reuses agg buffer as N*64)
    k_init_agg<<<2048, 256, 0, stream>>>(H, dinv, agg, 6, NN * 64);
    k_edge64<<<NE / 8, 256, 0, stream>>>(row, col, w, dinv, H, agg);

    // 7) GRU blend
    k_final<<<2048, 256, 0, stream>>>(agg, bh, z, hid, out);
}